// RelativePos_84232898609394
// MI455X (gfx1250) — compile-verified
//
#include <hip/hip_runtime.h>
#include <math.h>

typedef __attribute__((ext_vector_type(2))) float v2f;
typedef __attribute__((ext_vector_type(8))) float v8f;

#define B_    4
#define N_    4096
#define KNN_  16
#define EPS_  1e-5f

__device__ __forceinline__ float gelu_exact(float x) {
    return 0.5f * x * (1.0f + erff(x * 0.70710678118654752f));
}

__device__ __forceinline__ v8f wmma_f32(v2f a, v2f b, v8f c) {
    return __builtin_amdgcn_wmma_f32_16x16x4_f32(
        /*neg_a=*/false, a, /*neg_b=*/false, b,
        /*c_mod=*/(short)0, c, /*reuse_a=*/false, /*reuse_b=*/false);
}

// ---------------------------------------------------------------------------
// Kernel 1: brute-force KNN (top-16 by squared distance, self included).
// One thread per query point; wave-uniform candidate loads hit WGP$/L2.
// ---------------------------------------------------------------------------
__global__ __launch_bounds__(256) void knn_kernel(const float* __restrict__ xyz,
                                                  int* __restrict__ knn_idx) {
    int t = blockIdx.x * blockDim.x + threadIdx.x;   // 0..B*N-1
    int b = t >> 12;
    int n = t & (N_ - 1);
    const float* X = xyz + (size_t)b * 3 * N_;
    float qx = X[n], qy = X[N_ + n], qz = X[2 * N_ + n];

    float dk[KNN_];
    int   ik[KNN_];
#pragma unroll
    for (int s = 0; s < KNN_; ++s) { dk[s] = 3.4e38f; ik[s] = 0; }

    for (int j = 0; j < N_; ++j) {
        float dx = X[j] - qx;
        float dy = X[N_ + j] - qy;
        float dz = X[2 * N_ + j] - qz;
        float d = dx * dx + dy * dy + dz * dz;
        if (d < dk[KNN_ - 1]) {           // early-out: mostly false once warm
            float cd = d; int ci = j;
#pragma unroll
            for (int s = 0; s < KNN_; ++s) {  // sorted-insert compare/swap chain
                bool lt = cd < dk[s];
                float td = lt ? dk[s] : cd;
                dk[s]    = lt ? cd : dk[s];
                cd = td;
                int ti = lt ? ik[s] : ci;
                ik[s]  = lt ? ci : ik[s];
                ci = ti;
            }
        }
    }
#pragma unroll
    for (int s = 0; s < KNN_; ++s) knn_idx[(size_t)t * KNN_ + s] = ik[s];
}

// ---------------------------------------------------------------------------
// Kernel 2: fused geo -> MLP(6->128) -> MLP(128->128) -> MLP(128->768) -> max.
// One wave32 handles TWO points (two 16-row M-tiles sharing every B fragment:
// halves L2 weight traffic per FLOP). A-fragments live in 128 VGPRs so the
// hot loop is one global_load_b64 of weights + two v_wmma per K-step.
// ---------------------------------------------------------------------------
#define WPB 2   // waves per block
#define PPW 2   // points per wave

__global__ __launch_bounds__(WPB * 32) void mlp_kernel(
    const float* __restrict__ xyz, const int* __restrict__ knn_idx,
    const float* __restrict__ W1, const float* __restrict__ b1,
    const float* __restrict__ g1, const float* __restrict__ be1,
    const float* __restrict__ m1, const float* __restrict__ v1,
    const float* __restrict__ W2, const float* __restrict__ b2,
    const float* __restrict__ g2, const float* __restrict__ be2,
    const float* __restrict__ m2, const float* __restrict__ v2,
    const float* __restrict__ W3, const float* __restrict__ b3,
    float* __restrict__ out) {

    __shared__ float W1p[128 * 8];                 // W1 zero-padded 6->8 (4 KB)
    __shared__ float geo_s[WPB][PPW][16 * 8];      // per point: 16 x 8
    __shared__ float hbuf[WPB][PPW][16 * 132];     // per point: 16 x 128 (+4 pad)

    const int wave = threadIdx.x >> 5;
    const int lane = threadIdx.x & 31;
    const int pt0  = (blockIdx.x * WPB + wave) * PPW;   // first point of wave
    const int b    = pt0 >> 12;                          // PPW=2 -> same batch
    const int n0   = pt0 & (N_ - 1);
    const int n1   = n0 + 1;
    const float* X = xyz + (size_t)b * 3 * N_;

    // ---- stage W1 (zero-padded) into LDS: unconditional B-frag loads ----
    for (int i = threadIdx.x; i < 128 * 8; i += WPB * 32) {
        int row = i >> 3, col = i & 7;
        W1p[i] = (col < 6) ? W1[row * 6 + col] : 0.0f;
    }

    // ---- build geo: lanes 0-15 -> point0 rows, lanes 16-31 -> point1 rows ---
    {
        int p = lane >> 4, r = lane & 15;
        int pt = pt0 + p;
        int n = pt & (N_ - 1);
        int j = knn_idx[(size_t)pt * KNN_ + r];
        float cx = X[n], cy = X[N_ + n], cz = X[2 * N_ + n];
        float nx = X[j], ny = X[N_ + j], nz = X[2 * N_ + j];
        float* g = &geo_s[wave][p][r * 8];
        g[0] = nx - cx; g[1] = ny - cy; g[2] = nz - cz;
        g[3] = cx;      g[4] = cy;      g[5] = cz;
        g[6] = 0.0f;    g[7] = 0.0f;
    }
    __syncthreads();

    const int mrow  = lane & 15;        // A row / B column / D column index
    const int khalf = (lane >> 4) << 1; // 0 or 2 : K sub-pair per half-wave
    const int rbase = (lane >> 4) << 3; // D rows 0-7 or 8-15

    float* h0 = hbuf[wave][0];
    float* h1 = hbuf[wave][1];
    const float* ge0 = geo_s[wave][0];
    const float* ge1 = geo_s[wave][1];

    // ---- GEMM1: two 16x8 A-tiles @ 8x128, BN + GeLU, D-layout into hbuf ----
    {
        v2f a00, a01, a10, a11;
        a00.x = ge0[mrow * 8 + khalf];     a00.y = ge0[mrow * 8 + khalf + 1];
        a01.x = ge0[mrow * 8 + 4 + khalf]; a01.y = ge0[mrow * 8 + 4 + khalf + 1];
        a10.x = ge1[mrow * 8 + khalf];     a10.y = ge1[mrow * 8 + khalf + 1];
        a11.x = ge1[mrow * 8 + 4 + khalf]; a11.y = ge1[mrow * 8 + 4 + khalf + 1];
#pragma unroll
        for (int t = 0; t < 8; ++t) {
            int nc = t * 16 + mrow;
            v8f acc0 = {}, acc1 = {};
            v2f bb;
            bb.x = W1p[nc * 8 + khalf];     bb.y = W1p[nc * 8 + khalf + 1];
            acc0 = wmma_f32(a00, bb, acc0);
            acc1 = wmma_f32(a10, bb, acc1);
            bb.x = W1p[nc * 8 + 4 + khalf]; bb.y = W1p[nc * 8 + 4 + khalf + 1];
            acc0 = wmma_f32(a01, bb, acc0);
            acc1 = wmma_f32(a11, bb, acc1);

            float bias = b1[nc];
            float sc = g1[nc] * __frsqrt_rn(v1[nc] + EPS_);
            float mu = m1[nc], beta = be1[nc];
#pragma unroll
            for (int r = 0; r < 8; ++r) {
                float x0 = ((acc0[r] + bias) - mu) * sc + beta;
                float x1 = ((acc1[r] + bias) - mu) * sc + beta;
                h0[(r + rbase) * 132 + nc] = gelu_exact(x0);
                h1[(r + rbase) * 132 + nc] = gelu_exact(x1);
            }
        }
    }
    __syncthreads();

    // ---- hoist h1-stage A-fragments into registers (128 VGPRs) ----
    v2f af0[32], af1[32];
#pragma unroll
    for (int s = 0; s < 32; ++s) {
        int kb = s * 4 + khalf;
        af0[s].x = h0[mrow * 132 + kb]; af0[s].y = h0[mrow * 132 + kb + 1];
        af1[s].x = h1[mrow * 132 + kb]; af1[s].y = h1[mrow * 132 + kb + 1];
    }
    __syncthreads();

    // ---- GEMM2: 16x128 @ 128x128 (x2 points), BN + GeLU, write h2 to LDS ---
#pragma unroll
    for (int t = 0; t < 8; ++t) {
        int nc = t * 16 + mrow;
        const float* w2row = W2 + (size_t)nc * 128;
        v8f acc0 = {}, acc1 = {};
#pragma unroll
        for (int s = 0; s < 32; ++s) {
            int kb = s * 4 + khalf;
            v2f bb; bb.x = w2row[kb]; bb.y = w2row[kb + 1];
            acc0 = wmma_f32(af0[s], bb, acc0);
            acc1 = wmma_f32(af1[s], bb, acc1);
        }
        float bias = b2[nc];
        float sc = g2[nc] * __frsqrt_rn(v2[nc] + EPS_);
        float mu = m2[nc], beta = be2[nc];
#pragma unroll
        for (int r = 0; r < 8; ++r) {
            float x0 = ((acc0[r] + bias) - mu) * sc + beta;
            float x1 = ((acc1[r] + bias) - mu) * sc + beta;
            h0[(r + rbase) * 132 + nc] = gelu_exact(x0);   // afrag already in regs
            h1[(r + rbase) * 132 + nc] = gelu_exact(x1);
        }
    }
    __syncthreads();

    // ---- hoist h2-stage A-fragments ----
#pragma unroll
    for (int s = 0; s < 32; ++s) {
        int kb = s * 4 + khalf;
        af0[s].x = h0[mrow * 132 + kb]; af0[s].y = h0[mrow * 132 + kb + 1];
        af1[s].x = h1[mrow * 132 + kb]; af1[s].y = h1[mrow * 132 + kb + 1];
    }

    // ---- GEMM3: 16x128 @ 128x768 (x2 points), fused row-max, + b3 ----
    float* outb = out + (size_t)b * 768 * N_;
#pragma unroll 2
    for (int t = 0; t < 48; ++t) {
        const float* w3row = W3 + (size_t)(t * 16 + mrow) * 128;
        v8f acc0 = {}, acc1 = {};
#pragma unroll
        for (int s = 0; s < 32; ++s) {
            int kb = s * 4 + khalf;
            v2f bb; bb.x = w3row[kb]; bb.y = w3row[kb + 1];
            acc0 = wmma_f32(af0[s], bb, acc0);
            acc1 = wmma_f32(af1[s], bb, acc1);
        }
        float mx0 = acc0[0], mx1 = acc1[0];
#pragma unroll
        for (int r = 1; r < 8; ++r) {
            mx0 = fmaxf(mx0, acc0[r]);
            mx1 = fmaxf(mx1, acc1[r]);
        }
        mx0 = fmaxf(mx0, __shfl_xor(mx0, 16, 32));   // rows 0-7 vs 8-15
        mx1 = fmaxf(mx1, __shfl_xor(mx1, 16, 32));
        int c = t * 16 + mrow;
        float bias = b3[c];
        if (lane < 16) outb[(size_t)c * N_ + n0] = mx0 + bias;  // point0
        else           outb[(size_t)c * N_ + n1] = mx1 + bias;  // point1
    }
}

// ---------------------------------------------------------------------------
extern "C" void kernel_launch(void* const* d_in, const int* in_sizes, int n_in,
                              void* d_out, int out_size, void* d_ws, size_t ws_size,
                              hipStream_t stream) {
    const float* xyz = (const float*)d_in[0];
    const float* W1  = (const float*)d_in[1];
    const float* b1  = (const float*)d_in[2];
    const float* g1  = (const float*)d_in[3];
    const float* be1 = (const float*)d_in[4];
    const float* m1  = (const float*)d_in[5];
    const float* v1  = (const float*)d_in[6];
    const float* W2  = (const float*)d_in[7];
    const float* b2  = (const float*)d_in[8];
    const float* g2  = (const float*)d_in[9];
    const float* be2 = (const float*)d_in[10];
    const float* m2  = (const float*)d_in[11];
    const float* v2  = (const float*)d_in[12];
    const float* W3  = (const float*)d_in[13];
    const float* b3  = (const float*)d_in[14];
    float* out = (float*)d_out;

    int* knn_idx = (int*)d_ws;   // B*N*K ints = 1 MiB

    const int total_pts = B_ * N_;
    knn_kernel<<<total_pts / 256, 256, 0, stream>>>(xyz, knn_idx);
    mlp_kernel<<<total_pts / (WPB * PPW), WPB * 32, 0, stream>>>(
        xyz, knn_idx, W1, b1, g1, be1, m1, v1,
        W2, b2, g2, be2, m2, v2, W3, b3, out);
}